// GLVQ_51745765982761
// MI455X (gfx1250) — compile-verified
//
#include <hip/hip_runtime.h>
#include <math.h>

// GLVQ loss on MI455X (gfx1250), wave32 + WMMA fp32 path.
// N=8192, P=128, D=256, NUM_CLASSES=128.

#define GN 8192
#define GP 128
#define GD 256
#define KC 64          // K-chunk of prototypes staged in LDS
#define XS_LD 260      // padded row stride (floats) for x tile   (16B-aligned rows, bank-spread)
#define PS_LD 68       // padded row stride for proto K-chunk
#define DIST_LD 132    // padded row stride for dist tile
#define NBLK (GN / 16) // 512 blocks, 16 x-rows each

typedef __attribute__((ext_vector_type(2))) float v2f;
typedef __attribute__((ext_vector_type(8))) float v8f;

// ---------------------------------------------------------------------------
// Kernel 1: p_sq[p] = sum_d proto[p][d]^2
// ---------------------------------------------------------------------------
__global__ void glvq_psq(const float* __restrict__ proto, float* __restrict__ psq) {
    int p = threadIdx.x;
    if (p < GP) {
        float s = 0.f;
        for (int d = 0; d < GD; ++d) {
            float v = proto[p * GD + d];
            s = fmaf(v, v, s);
        }
        psq[p] = s;
    }
}

// ---------------------------------------------------------------------------
// Kernel 2: per-block (16 rows x 128 protos) distances via v_wmma_f32_16x16x4,
// then in-block d1/d2 reduction -> sigmoid -> one partial sum per block.
// ---------------------------------------------------------------------------
__global__ __launch_bounds__(256)
void glvq_main(const float* __restrict__ x, const int* __restrict__ y,
               const float* __restrict__ proto, const float* __restrict__ psq,
               float* __restrict__ partials) {
    __shared__ float xs[16 * XS_LD];       // x tile, 16 x 256 (padded)
    __shared__ float ps[GP * PS_LD];       // proto K-chunk, 128 x 64 (padded)
    __shared__ float dist[16 * DIST_LD];   // -2*dot tile, 16 x 128 (padded)
    __shared__ float psq_l[GP];
    __shared__ float xsq_part[16][8];
    __shared__ float rmin[16][8];
    __shared__ float rd1[16][8];
    __shared__ float sig[16];

    const int tid  = threadIdx.x;
    const int lane = tid & 31;
    const int wave = tid >> 5;             // 8 waves
    const int row0 = blockIdx.x * 16;

    if (tid < GP) psq_l[tid] = psq[tid];

    // Stage x tile: 16 rows are contiguous in memory -> 1024 coalesced float4s.
    {
        const float4* xg = (const float4*)(x + (size_t)row0 * GD);
#pragma unroll
        for (int i = 0; i < 4; ++i) {
            int f4i = tid + i * 256;       // 64 float4 per row
            int r = f4i >> 6;
            int c = (f4i & 63) << 2;
            float4 v = xg[f4i];
            *(float4*)&xs[r * XS_LD + c] = v;
        }
    }

    const int m    = lane & 15;            // row / proto index within fragment
    const int half = lane >> 4;            // lane half selects K pair
    const int p0   = wave * 16;            // this wave's 16 prototypes

    v8f acc = {0.f, 0.f, 0.f, 0.f, 0.f, 0.f, 0.f, 0.f};

    for (int kc = 0; kc < GD; kc += KC) {
        __syncthreads();                   // prev chunk fully consumed (and xs ready)
        // Stage proto K-chunk: 128 rows x 64 floats = 2048 float4s.
#pragma unroll
        for (int i = 0; i < 8; ++i) {
            int f4i = tid + i * 256;       // 16 float4 per chunk-row
            int r = f4i >> 4;
            int c = (f4i & 15) << 2;
            float4 v = *(const float4*)(proto + (size_t)r * GD + kc + c);
            *(float4*)&ps[r * PS_LD + c] = v;
        }
        __syncthreads();

        const float* xrow = &xs[m * XS_LD + kc];
        const float* prow = &ps[(p0 + m) * PS_LD];
#pragma unroll
        for (int k4 = 0; k4 < KC; k4 += 4) {
            // A (16x4 f32): lane L holds row m = L%16; lanes 0-15 K={0,1} in
            // VGPR0/1, lanes 16-31 K={2,3}  -> contiguous float2 at k4+2*half.
            v2f a = *(const v2f*)&xrow[k4 + 2 * half];
            // B (4x16 f32): symmetric layout (K-range split across lane
            // halves, consecutive K across the 2 VGPRs within a half):
            // lane L holds col m = L%16, K = k4 + 2*half + {0,1}.
            v2f b = *(const v2f*)&prow[k4 + 2 * half];
            acc = __builtin_amdgcn_wmma_f32_16x16x4_f32(
                false, a, false, b, (short)0, acc, false, false);
        }
    }

    // Spill -2*dot tiles. C/D layout: VGPR i -> (M = i + 8*half, N = m).
#pragma unroll
    for (int i = 0; i < 8; ++i) {
        dist[(i + 8 * half) * DIST_LD + p0 + m] = -2.0f * acc[i];
    }

    // x_sq partials from the resident x tile.
    if (tid < 128) {
        int r = tid >> 3, c = (tid & 7) * 32;
        float s = 0.f;
        for (int j = 0; j < 32; ++j) {
            float v = xs[r * XS_LD + c + j];
            s = fmaf(v, v, s);
        }
        xsq_part[r][tid & 7] = s;
    }
    __syncthreads();

    // Per-row scan: d1 (p == y[r]) and min over wrong prototypes, 8 segments/row.
    if (tid < 128) {
        int r = tid >> 3, seg = (tid & 7) * 16;
        int yr = y[row0 + r];
        float mn = INFINITY, dv = INFINITY;
        for (int j = 0; j < 16; ++j) {
            int p = seg + j;
            float v = dist[r * DIST_LD + p] + psq_l[p];
            if (p == yr) dv = v;
            else         mn = fminf(mn, v);
        }
        rmin[r][tid & 7] = mn;
        rd1[r][tid & 7]  = dv;
    }
    __syncthreads();

    if (tid < 16) {
        float xq = 0.f, mn = INFINITY, dv = INFINITY;
#pragma unroll
        for (int j = 0; j < 8; ++j) {
            xq += xsq_part[tid][j];
            mn = fminf(mn, rmin[tid][j]);
            dv = fminf(dv, rd1[tid][j]);
        }
        float d1 = xq + dv;
        float d2 = xq + mn;
        float mu = (d1 - d2) / (d1 + d2);
        sig[tid] = 1.0f / (1.0f + __expf(mu));   // sigmoid(-mu)
    }
    __syncthreads();

    if (tid == 0) {
        float s = 0.f;
#pragma unroll
        for (int j = 0; j < 16; ++j) s += sig[j];
        partials[blockIdx.x] = s;
    }
}

// ---------------------------------------------------------------------------
// Kernel 3: deterministic fixed-order reduction of the 512 block partials.
// ---------------------------------------------------------------------------
__global__ void glvq_final(const float* __restrict__ partials, float* __restrict__ out) {
    if (threadIdx.x == 0) {
        float s = 0.f;
        for (int i = 0; i < NBLK; ++i) s += partials[i];
        out[0] = s * (1.0f / (float)GN);
    }
}

// ---------------------------------------------------------------------------
extern "C" void kernel_launch(void* const* d_in, const int* in_sizes, int n_in,
                              void* d_out, int out_size, void* d_ws, size_t ws_size,
                              hipStream_t stream) {
    const float* x     = (const float*)d_in[0];
    const int*   y     = (const int*)d_in[1];
    const float* proto = (const float*)d_in[2];

    float* ws       = (float*)d_ws;
    float* psq      = ws;          // [128]
    float* partials = ws + 128;    // [512]

    glvq_psq<<<1, 128, 0, stream>>>(proto, psq);
    glvq_main<<<NBLK, 256, 0, stream>>>(x, y, proto, psq, partials);
    glvq_final<<<1, 32, 0, stream>>>(partials, (float*)d_out);
}